// TopKTokenChoiceRouter_2302102471508
// MI455X (gfx1250) — compile-verified
//
#include <hip/hip_runtime.h>
#include <math.h>
#include <stdint.h>

typedef __attribute__((ext_vector_type(2))) float v2f;
typedef __attribute__((ext_vector_type(8))) float v8f;

#define TOKENS 8192
#define HS     4096
#define NE     64
#define KC     64
#define TPB    32    // tokens per block
#define LDST   68    // 64 + 4 pad: 16B-aligned rows, conflict-free b64 fragment reads

__global__ __launch_bounds__(256)
void router_topk_kernel(const float* __restrict__ x, const float* __restrict__ W,
                        float* __restrict__ out_w, int* __restrict__ out_i)
{
    __shared__ float xs[2][TPB * LDST];   // ping-pong: 32 tokens  x KC
    __shared__ float ws[2][NE  * LDST];   // ping-pong: 64 experts x KC
    __shared__ float lg[TPB * LDST];      // 32 tokens x 64 logits

    const int tid  = threadIdx.x;
    const int wave = tid >> 5;              // 0..7
    const int lane = tid & 31;
    const int half = lane >> 4;             // 0: K+{0,1}   1: K+{2,3}
    const int lm   = lane & 15;

    const int ttile = (wave >> 2) * 16;     // token tile base in block (0 / 16)
    const int ebase = (wave & 3) * 16;      // expert tile base (0,16,32,48)
    const int tok0  = blockIdx.x * TPB;

    // ---- per-thread async-copy addresses: 2 x-vec4 + 4 W-vec4 per thread ----
    uint64_t gx[2]; uint32_t lox[2];
    uint64_t gw[4]; uint32_t low[4];
    #pragma unroll
    for (int r = 0; r < 2; ++r) {           // 512 float4 slots: 32 rows x 16
        int idx = tid + r * 256, row = idx >> 4, c4 = idx & 15;
        gx[r]  = (uint64_t)(uintptr_t)(x + (size_t)(tok0 + row) * HS + c4 * 4);
        lox[r] = (uint32_t)(row * LDST * 4 + c4 * 16);
    }
    #pragma unroll
    for (int r = 0; r < 4; ++r) {           // 1024 float4 slots: 64 rows x 16
        int idx = tid + r * 256, row = idx >> 4, c4 = idx & 15;
        gw[r]  = (uint64_t)(uintptr_t)(W + (size_t)row * HS + c4 * 4);
        low[r] = (uint32_t)(row * LDST * 4 + c4 * 16);
    }
    // flat->LDS: low 32 bits of a __shared__ flat address = workgroup LDS offset
    const uint32_t xsb[2] = { (uint32_t)(uintptr_t)&xs[0][0], (uint32_t)(uintptr_t)&xs[1][0] };
    const uint32_t wsb[2] = { (uint32_t)(uintptr_t)&ws[0][0], (uint32_t)(uintptr_t)&ws[1][0] };

    v8f acc = {};

    // ---- prologue: async-copy chunk 0 into buffer 0 (no VGPR data path) ----
    #pragma unroll
    for (int r = 0; r < 2; ++r)
        asm volatile("global_load_async_to_lds_b128 %0, %1, off"
                     :: "v"(xsb[0] + lox[r]), "v"(gx[r]) : "memory");
    #pragma unroll
    for (int r = 0; r < 4; ++r)
        asm volatile("global_load_async_to_lds_b128 %0, %1, off"
                     :: "v"(wsb[0] + low[r]), "v"(gw[r]) : "memory");

    int cur = 0;
    for (int kc = 0; kc < HS; kc += KC, cur ^= 1) {
        // current buffer complete (own copies), then visible across all waves
        asm volatile("s_wait_asynccnt 0x0" ::: "memory");
        __syncthreads();

        // issue next chunk's async copies into the other buffer; they fly
        // during the WMMA block below (no wait here)
        if (kc + KC < HS) {
            const uint64_t kb  = (uint64_t)(kc + KC) * 4;
            const int      nxt = cur ^ 1;
            #pragma unroll
            for (int r = 0; r < 2; ++r)
                asm volatile("global_load_async_to_lds_b128 %0, %1, off"
                             :: "v"(xsb[nxt] + lox[r]), "v"(gx[r] + kb) : "memory");
            #pragma unroll
            for (int r = 0; r < 4; ++r)
                asm volatile("global_load_async_to_lds_b128 %0, %1, off"
                             :: "v"(wsb[nxt] + low[r]), "v"(gw[r] + kb) : "memory");
        }

        // ---- 16 fp32 WMMA k-steps from the current buffer ----
        const float* xsc = &xs[cur][0];
        const float* wsc = &ws[cur][0];
        #pragma unroll
        for (int k = 0; k < KC; k += 4) {
            // A 16x4: lane(half,lm) = x[token ttile+lm][k + 2*half .. +1]
            v2f a = *reinterpret_cast<const v2f*>(&xsc[(ttile + lm) * LDST + k + 2 * half]);
            // B 4x16: lane(half,lm) = W[expert ebase+lm][k + 2*half .. +1]
            v2f b = *reinterpret_cast<const v2f*>(&wsc[(ebase + lm) * LDST + k + 2 * half]);
            acc = __builtin_amdgcn_wmma_f32_16x16x4_f32(
                false, a, false, b, (short)0, acc, false, false);
        }
        // no trailing barrier: next iteration's wait+barrier precedes any
        // write into the buffer this iteration just read
    }

    // ---- scatter D tile: lane holds column N=ebase+lm, rows M=v+8*half ----
    #pragma unroll
    for (int v = 0; v < 8; ++v) {
        int m = ttile + v + 8 * half;
        lg[m * LDST + ebase + lm] = acc[v];
    }
    __syncthreads();

    // ---- exact fp32 softmax + top-2, one thread per token ----
    if (tid < TPB) {
        const float* row = &lg[tid * LDST];
        float m1 = -INFINITY, m2 = -INFINITY;
        int   i1 = 0, i2 = 0;
        for (int e = 0; e < NE; ++e) {
            float l = row[e];
            if (l > m1)      { m2 = m1; i2 = i1; m1 = l; i1 = e; }
            else if (l > m2) { m2 = l;  i2 = e; }
        }
        float z = 0.f;
        for (int e = 0; e < NE; ++e) z += expf(row[e] - m1);
        float invz = 1.f / z;
        int gt = tok0 + tid;
        out_w[gt * 2 + 0] = invz;                  // exp(m1-m1)/Z
        out_w[gt * 2 + 1] = expf(m2 - m1) * invz;  // exp(m2-m1)/Z
        out_i[gt * 2 + 0] = i1;
        out_i[gt * 2 + 1] = i2;
    }
}

extern "C" void kernel_launch(void* const* d_in, const int* in_sizes, int n_in,
                              void* d_out, int out_size, void* d_ws, size_t ws_size,
                              hipStream_t stream) {
    const float* x = (const float*)d_in[0];   // (2048, 4, 4096) fp32 -> 8192 x 4096
    const float* W = (const float*)d_in[1];   // (64, 4096) fp32
    float* out_w = (float*)d_out;                      // weights: 8192 x 2 fp32
    int*   out_i = (int*)d_out + (size_t)TOKENS * 2;   // indices: 8192 x 2 i32 (raw bits)
    router_topk_kernel<<<TOKENS / TPB, 256, 0, stream>>>(x, W, out_w, out_i);
}